// TtLlamaRotary_45964740002444
// MI455X (gfx1250) — compile-verified
//
#include <hip/hip_runtime.h>
#include <hip/hip_bf16.h>
#include <stdint.h>

// ---------------------------------------------------------------------------
// Fused interleaved RoPE for Q and K (MI455X / gfx1250).
//
// out[..,2i]   = x[..,2i]  *cos[2i]   - x[..,2i+1]*sin[2i]
// out[..,2i+1] = x[..,2i+1]*cos[2i+1] + x[..,2i]  *sin[2i+1]
//
// Memory-bound (0.5 FLOP/B): strategy is pure streaming with NT hints for
// x/out, plus TDM (tensor_load_to_lds) staging of the per-(s,d) cos/sin tile
// into LDS, reused across 8 heads per block to cut L2 read amplification.
// ---------------------------------------------------------------------------

typedef float    f32x4 __attribute__((ext_vector_type(4)));
typedef uint32_t u32x4 __attribute__((ext_vector_type(4)));
typedef int      i32x4 __attribute__((ext_vector_type(4)));
typedef int      i32x8 __attribute__((ext_vector_type(8)));

#define ROWS     16                  // seq rows per block chunk
#define DHEAD    128
#define CHUNK    (ROWS * DHEAD)      // 2048 floats = 8 KB per table
#define THREADS  256                 // 8 wave32 per block
#define SEQ      4096
#define NB       4
#define NHQ      32
#define NHK      8

// Issue a 1-D TDM copy of `nelem` f32 from global `gaddr` into LDS byte
// address `lds_byte`. Wave-level instruction; call from one wave only.
__device__ __forceinline__ void tdm_load_1d(uint32_t lds_byte, const void* gaddr,
                                            uint32_t nelem)
{
#if __has_builtin(__builtin_amdgcn_tensor_load_to_lds)
    const uint64_t ga = (uint64_t)(uintptr_t)gaddr;
    u32x4 g0;
    g0[0] = 1u;                                            // count=1, user mode
    g0[1] = lds_byte;                                      // lds_addr
    g0[2] = (uint32_t)ga;                                  // global_addr[31:0]
    g0[3] = (uint32_t)((ga >> 32) & 0x1FFFFFFu)            // global_addr[56:32]
          | (2u << 30);                                    // type = 2 (image)
    i32x8 g1;
    g1[0] = 0x00020000;                                    // data_size = 4B
    g1[1] = (int)((nelem & 0xFFFFu) << 16);                // tensor_dim0[15:0]
    g1[2] = (int)((nelem >> 16) | (1u << 16));             // dim0[31:16], tensor_dim1=1
    g1[3] = (int)((nelem & 0xFFFFu) << 16);                // tile_dim0 = nelem
    g1[4] = 1;                                             // tile_dim1 = 1
    g1[5] = (int)nelem;                                    // tensor_dim0_stride
    g1[6] = 0;
    g1[7] = 0;
    i32x4 gz = {0, 0, 0, 0};
#if defined(__clang_major__) && (__clang_major__ >= 23)
    i32x8 gz8 = {0, 0, 0, 0, 0, 0, 0, 0};
    __builtin_amdgcn_tensor_load_to_lds(g0, g1, gz, gz, gz8, 0);
#else
    __builtin_amdgcn_tensor_load_to_lds(g0, g1, gz, gz, 0);
#endif
#endif
}

__device__ __forceinline__ void wait_tensorcnt0()
{
#if __has_builtin(__builtin_amdgcn_s_wait_tensorcnt)
    __builtin_amdgcn_s_wait_tensorcnt(0);
#else
    asm volatile("s_wait_tensorcnt 0x0" ::: "memory");
#endif
}

__global__ void __launch_bounds__(THREADS)
rope_qk_kernel(const float* __restrict__ xq, const float* __restrict__ xk,
               const float* __restrict__ cost, const float* __restrict__ sint,
               float* __restrict__ out)
{
    __shared__ float lds_cos[CHUNK];
    __shared__ float lds_sin[CHUNK];

    const int chunk = blockIdx.x;          // 0..SEQ/ROWS-1
    const int b     = blockIdx.y;          // 0..3
    const int z     = blockIdx.z;          // 0..3 -> q head groups, 4 -> k
    const int s0    = chunk * ROWS;
    const size_t cs_off = (size_t)s0 * DHEAD;

    // ---- stage cos/sin tile into LDS ------------------------------------
#if __has_builtin(__builtin_amdgcn_tensor_load_to_lds)
    if (threadIdx.x < 32) {                // wave 0 only (uniform branch)
        const uint32_t lc = (uint32_t)(uintptr_t)&lds_cos[0]; // low32 = LDS offset
        const uint32_t ls = (uint32_t)(uintptr_t)&lds_sin[0];
        tdm_load_1d(lc, cost + cs_off, CHUNK);
        tdm_load_1d(ls, sint + cs_off, CHUNK);
        wait_tensorcnt0();
    }
#else
    for (int i = threadIdx.x; i < CHUNK; i += THREADS) {
        lds_cos[i] = cost[cs_off + i];
        lds_sin[i] = sint[cs_off + i];
    }
#endif
    __syncthreads();

    // ---- per-block head group -------------------------------------------
    const bool  is_k = (z == 4);
    const float* __restrict__ x  = is_k ? xk : xq;
    const int    nh  = is_k ? NHK : NHQ;
    const int    h0  = is_k ? 0 : z * 8;
    const size_t qsz = (size_t)NB * NHQ * SEQ * DHEAD;
    float* __restrict__ o = is_k ? (out + qsz) : out;

    const int t  = threadIdx.x;
    const int f1 = t * 4;              // floats [0, 1024)
    const int f2 = f1 + CHUNK / 2;     // floats [1024, 2048)

    // cos/sin registers reused across all 8 head iterations
    const f32x4 cv1 = *(const f32x4*)&lds_cos[f1];
    const f32x4 sv1 = *(const f32x4*)&lds_sin[f1];
    const f32x4 cv2 = *(const f32x4*)&lds_cos[f2];
    const f32x4 sv2 = *(const f32x4*)&lds_sin[f2];

#pragma unroll
    for (int hi = 0; hi < 8; ++hi) {
        const int    h    = h0 + hi;
        const size_t base = (((size_t)b * nh + h) * SEQ + (size_t)s0) * DHEAD;

        const f32x4 x1 = __builtin_nontemporal_load((const f32x4*)(x + base + f1));
        const f32x4 x2 = __builtin_nontemporal_load((const f32x4*)(x + base + f2));

        f32x4 o1, o2;
        o1.x = x1.x * cv1.x - x1.y * sv1.x;
        o1.y = x1.y * cv1.y + x1.x * sv1.y;
        o1.z = x1.z * cv1.z - x1.w * sv1.z;
        o1.w = x1.w * cv1.w + x1.z * sv1.w;

        o2.x = x2.x * cv2.x - x2.y * sv2.x;
        o2.y = x2.y * cv2.y + x2.x * sv2.y;
        o2.z = x2.z * cv2.z - x2.w * sv2.z;
        o2.w = x2.w * cv2.w + x2.z * sv2.w;

        __builtin_nontemporal_store(o1, (f32x4*)(o + base + f1));
        __builtin_nontemporal_store(o2, (f32x4*)(o + base + f2));
    }
}

extern "C" void kernel_launch(void* const* d_in, const int* in_sizes, int n_in,
                              void* d_out, int out_size, void* d_ws, size_t ws_size,
                              hipStream_t stream)
{
    (void)in_sizes; (void)n_in; (void)out_size; (void)d_ws; (void)ws_size;
    const float* xq   = (const float*)d_in[0];
    const float* xk   = (const float*)d_in[1];
    const float* cost = (const float*)d_in[2];
    const float* sint = (const float*)d_in[3];
    // d_in[4] = trans_mat: its effect (interleaved rotate_half) is applied
    // analytically in-kernel; no need to load it.
    float* out = (float*)d_out;

    dim3 grid(SEQ / ROWS, NB, 5);   // z: 4 groups of 8 q-heads + 1 group of 8 k-heads
    dim3 block(THREADS);
    rope_qk_kernel<<<grid, block, 0, stream>>>(xq, xk, cost, sint, out);
}